// ExpectedSignature_40621800686198
// MI455X (gfx1250) — compile-verified
//
#include <hip/hip_runtime.h>

// Problem constants (from reference): path (64, 256, 128, 5) f32 -> out (64, 780) f32
#define DIM     5
#define LPTS    128
#define NPATH   16384     // 64 * 256
#define NSAMP   256
#define NBATCH  64
#define S_TOTAL 780
#define SPAD    784       // 49 tiles of 16, 16B-aligned rows
#define STILES  49

typedef __attribute__((ext_vector_type(2))) float v2f;
typedef __attribute__((ext_vector_type(8))) float v8f;

// ---------------------------------------------------------------------------
// Kernel 1: per-lane sequential Chen recursion, full signature state in VGPRs.
// One path per lane; 64-thread blocks (2 waves). waves_per_eu(1) frees the
// full 1024-VGPR budget so s1..s4 (780 f32) + temps stay register-resident.
// ---------------------------------------------------------------------------
template <bool USE_WS>
__global__ __launch_bounds__(64)
__attribute__((amdgpu_waves_per_eu(1)))
void sig_kernel(const float* __restrict__ path, float* __restrict__ Y,
                float* __restrict__ out)
{
    const int p = blockIdx.x * 64 + threadIdx.x;        // path id, exact grid
    const float* __restrict__ row = path + (size_t)p * (LPTS * DIM);

    float s1[DIM];
    float s2[DIM][DIM];
    float s3[DIM][DIM][DIM];
    float s4[DIM][DIM][DIM][DIM];
#pragma unroll
    for (int a = 0; a < DIM; ++a) {
        s1[a] = 0.f;
#pragma unroll
        for (int b = 0; b < DIM; ++b) {
            s2[a][b] = 0.f;
#pragma unroll
            for (int c = 0; c < DIM; ++c) {
                s3[a][b][c] = 0.f;
#pragma unroll
                for (int d = 0; d < DIM; ++d) s4[a][b][c][d] = 0.f;
            }
        }
    }

    float xp[DIM];
#pragma unroll
    for (int i = 0; i < DIM; ++i) xp[i] = row[i];

#pragma unroll 1
    for (int t = 1; t < LPTS; ++t) {
        if (t + 6 < LPTS) __builtin_prefetch(row + (t + 6) * DIM, 0, 0);

        float v[DIM];
#pragma unroll
        for (int i = 0; i < DIM; ++i) {
            float xn = row[t * DIM + i];
            v[i] = xn - xp[i];
            xp[i] = xn;
        }

        // e2 = v (x) v / 2
        float e2[DIM][DIM];
#pragma unroll
        for (int a = 0; a < DIM; ++a)
#pragma unroll
            for (int b = 0; b < DIM; ++b) e2[a][b] = 0.5f * v[a] * v[b];

        float v12[DIM], s1a3[DIM];
#pragma unroll
        for (int i = 0; i < DIM; ++i) {
            v12[i]  = v[i] * (1.f / 12.f);   // e4 = e3*v/4 fold
            s1a3[i] = s1[i] * (1.f / 3.f);   // c1 (x) e3 fold (e3 = e2*v/3)
        }

        // Level 4: s4 += e4 + c1(x)e3 + c2(x)e2 + c3(x)v   (old carries)
        //   = (s3[abc] + e2[ab]*v[c]/12 + s1[a]/3 * e2[bc]) * v[d] + s2[ab]*e2[cd]
#pragma unroll
        for (int a = 0; a < DIM; ++a)
#pragma unroll
            for (int b = 0; b < DIM; ++b)
#pragma unroll
                for (int c = 0; c < DIM; ++c) {
                    float t2 = s3[a][b][c];
                    t2 = __builtin_fmaf(e2[a][b], v12[c], t2);
                    t2 = __builtin_fmaf(s1a3[a], e2[b][c], t2);
#pragma unroll
                    for (int d = 0; d < DIM; ++d) {
                        float acc = s4[a][b][c][d];
                        acc = __builtin_fmaf(s2[a][b], e2[c][d], acc);
                        acc = __builtin_fmaf(t2, v[d], acc);
                        s4[a][b][c][d] = acc;
                    }
                }

        // Level 3: s3 += e3 + c1(x)e2 + c2(x)v  = (e2[ab]/3 + s2[ab])*v[c] + s1[a]*e2[bc]
#pragma unroll
        for (int a = 0; a < DIM; ++a)
#pragma unroll
            for (int b = 0; b < DIM; ++b) {
                float tt = __builtin_fmaf(e2[a][b], (1.f / 3.f), s2[a][b]);
#pragma unroll
                for (int c = 0; c < DIM; ++c) {
                    float acc = s3[a][b][c];
                    acc = __builtin_fmaf(s1[a], e2[b][c], acc);
                    acc = __builtin_fmaf(tt, v[c], acc);
                    s3[a][b][c] = acc;
                }
            }

        // Level 2: s2 += e2 + c1(x)v
#pragma unroll
        for (int a = 0; a < DIM; ++a)
#pragma unroll
            for (int b = 0; b < DIM; ++b)
                s2[a][b] = __builtin_fmaf(s1[a], v[b], s2[a][b] + e2[a][b]);

        // Level 1
#pragma unroll
        for (int a = 0; a < DIM; ++a) s1[a] += v[a];
    }

    // ---- per-level squared norms ----
    float n1 = 0.f, n2 = 0.f, n3 = 0.f, n4 = 0.f;
#pragma unroll
    for (int a = 0; a < DIM; ++a) {
        n1 = __builtin_fmaf(s1[a], s1[a], n1);
#pragma unroll
        for (int b = 0; b < DIM; ++b) {
            n2 = __builtin_fmaf(s2[a][b], s2[a][b], n2);
#pragma unroll
            for (int c = 0; c < DIM; ++c) {
                n3 = __builtin_fmaf(s3[a][b][c], s3[a][b][c], n3);
#pragma unroll
                for (int d = 0; d < DIM; ++d)
                    n4 = __builtin_fmaf(s4[a][b][c][d], s4[a][b][c][d], n4);
            }
        }
    }

    // ---- dilatation: phi (C=4) + 60-step bisection on monotone polynomial ----
    float nq  = 1.f + n1 + n2 + n3 + n4;
    float phi = (nq > 4.f) ? (8.f - 16.f / nq) : nq;
    float c0  = 1.f - phi;

    float lo = 0.f, hi = 2.f;
#pragma unroll 1
    for (int it = 0; it < 60; ++it) {
        float mid = 0.5f * (lo + hi);
        float m2  = mid * mid;
        float f   = c0 + m2 * (n1 + m2 * (n2 + m2 * (n3 + m2 * n4)));
        if (f > 0.f) hi = mid; else lo = mid;
    }
    float lam = fminf(0.5f * (lo + hi), 1.f);
    float l1 = lam, l2 = lam * lam;
    float l3 = l2 * lam, l4 = l2 * l2;

    // ---- emit scaled signature ----
    float* __restrict__ yr = USE_WS ? (Y + (size_t)p * SPAD) : nullptr;
    float* __restrict__ ob = USE_WS ? nullptr : (out + (size_t)(p >> 8) * S_TOTAL);
    const float sc = 1.f / 256.f;

    auto emit = [&](int i, float val) {
        if constexpr (USE_WS) yr[i] = val;
        else                  atomicAdd(ob + i, val * sc);
    };

    int idx = 0;
#pragma unroll
    for (int a = 0; a < DIM; ++a) emit(idx++, l1 * s1[a]);
#pragma unroll
    for (int a = 0; a < DIM; ++a)
#pragma unroll
        for (int b = 0; b < DIM; ++b) emit(idx++, l2 * s2[a][b]);
#pragma unroll
    for (int a = 0; a < DIM; ++a)
#pragma unroll
        for (int b = 0; b < DIM; ++b)
#pragma unroll
            for (int c = 0; c < DIM; ++c) emit(idx++, l3 * s3[a][b][c]);
#pragma unroll
    for (int a = 0; a < DIM; ++a)
#pragma unroll
        for (int b = 0; b < DIM; ++b)
#pragma unroll
            for (int c = 0; c < DIM; ++c)
#pragma unroll
                for (int d = 0; d < DIM; ++d) emit(idx++, l4 * s4[a][b][c][d]);

    if constexpr (USE_WS) {           // zero pad columns 780..783 for the WMMA pass
        yr[780] = 0.f; yr[781] = 0.f; yr[782] = 0.f; yr[783] = 0.f;
    }
}

// ---------------------------------------------------------------------------
// Kernel 2: mean over N=256 as a segmented-sum GEMM on the matrix core.
// O[s, b] = sum_k A[s,k] * B[k,b],  A = Y rows (4 consecutive n per k-step),
// B = one-hot column of the owning batch. V_WMMA_F32_16X16X4_F32, f32 accum.
// One wave per (s-tile, 16-batch group); EXEC all ones as WMMA requires.
// ---------------------------------------------------------------------------
__global__ __launch_bounds__(32)
void reduce_kernel(const float* __restrict__ Y, float* __restrict__ out)
{
    const int stile = blockIdx.x % STILES;
    const int bg    = blockIdx.x / STILES;      // 0..3  (group of 16 batches)
    const int lane  = threadIdx.x;              // 0..31
    const int m     = lane & 15;                // A: M row (s), B/D: N col (b)
    const int kh    = lane >> 4;                // A holds K = {2*kh, 2*kh+1}

    const int s0 = stile * 16;
    v8f acc = {0.f, 0.f, 0.f, 0.f, 0.f, 0.f, 0.f, 0.f};

#pragma unroll 1
    for (int boff = 0; boff < 16; ++boff) {
        const float bv = (m == boff) ? 1.f : 0.f;   // one-hot, uniform over K
        v2f B; B.x = bv; B.y = bv;
        const float* __restrict__ col =
            Y + ((size_t)(bg * 16 + boff) * NSAMP) * SPAD + s0 + m;
#pragma unroll 4
        for (int c2 = 0; c2 < NSAMP / 4; ++c2) {    // 64 k-steps of K=4 n's
            const float* ap = col + (size_t)(c2 * 4 + 2 * kh) * SPAD;
            v2f A; A.x = ap[0]; A.y = ap[SPAD];
            acc = __builtin_amdgcn_wmma_f32_16x16x4_f32(
                /*neg_a=*/false, A, /*neg_b=*/false, B,
                /*c_mod=*/(short)0, acc, /*reuse_a=*/false, /*reuse_b=*/false);
        }
    }

    const int bcol = bg * 16 + m;                   // D: N = lane&15
#pragma unroll
    for (int r = 0; r < 8; ++r) {                   // D: M = r + 8*(lane>>4)
        int srow = s0 + r + 8 * kh;
        if (srow < S_TOTAL)
            out[(size_t)bcol * S_TOTAL + srow] = acc[r] * (1.f / 256.f);
    }
}

// ---------------------------------------------------------------------------
extern "C" void kernel_launch(void* const* d_in, const int* in_sizes, int n_in,
                              void* d_out, int out_size, void* d_ws, size_t ws_size,
                              hipStream_t stream)
{
    (void)in_sizes; (void)n_in;
    const float* path = (const float*)d_in[0];
    float* out = (float*)d_out;

    const size_t need = (size_t)NPATH * SPAD * sizeof(float);   // ~51 MB
    if (d_ws != nullptr && ws_size >= need) {
        float* Yst = (float*)d_ws;
        sig_kernel<true><<<dim3(NPATH / 64), dim3(64), 0, stream>>>(path, Yst, out);
        reduce_kernel<<<dim3(STILES * (NBATCH / 16)), dim3(32), 0, stream>>>(Yst, out);
    } else {
        // Fallback without staging buffer: zero output, atomically accumulate.
        hipMemsetAsync(out, 0, (size_t)out_size * sizeof(float), stream);
        sig_kernel<false><<<dim3(NPATH / 64), dim3(64), 0, stream>>>(path, nullptr, out);
    }
}